// Net_32238024524262
// MI455X (gfx1250) — compile-verified
//
#include <hip/hip_runtime.h>
#include <hip/hip_bf16.h>

#define NN    50000
#define EE    800000
#define IN_C  128
#define HID   256
#define OUT_C 51

typedef __attribute__((ext_vector_type(2))) float v2f;
typedef __attribute__((ext_vector_type(8))) float v8f;

// ---------------------------------------------------------------------------
// Zero-fill
// ---------------------------------------------------------------------------
__global__ __launch_bounds__(256) void fill_zero_kernel(float* __restrict__ p, long long n) {
    long long i = (long long)blockIdx.x * blockDim.x + threadIdx.x;
    if (i < n) p[i] = 0.0f;
}

// ---------------------------------------------------------------------------
// Edge scatter: agg[dst] += w_e * x[src].  F multiple of 4.
// One float4 chunk per thread; chunks of one edge are contiguous threads, so
// a wave (32 lanes) covers 128 floats of one edge -> fully coalesced gather
// and coalesced fp32 atomics (L2-resident working set).
// ---------------------------------------------------------------------------
__global__ __launch_bounds__(256) void scatter_add_kernel(
    const float* __restrict__ x, const long long* __restrict__ ei,
    const float* __restrict__ ew, float* __restrict__ agg,
    long long E, int F)
{
    long long tid = (long long)blockIdx.x * blockDim.x + threadIdx.x;
    int chunks = F >> 2;                         // float4 chunks per edge
    long long e = tid / chunks;
    int c = (int)(tid - e * (long long)chunks);
    if (e >= E) return;
    long long src = ei[e];
    long long dst = ei[E + e];
    float w = ew[e];
    const float4* xs = (const float4*)(x + src * (long long)F) + c;
    float4 v = *xs;
    float* ap = agg + dst * (long long)F + (long long)c * 4;
    unsafeAtomicAdd(ap + 0, v.x * w);
    unsafeAtomicAdd(ap + 1, v.y * w);
    unsafeAtomicAdd(ap + 2, v.z * w);
    unsafeAtomicAdd(ap + 3, v.w * w);
}

// ---------------------------------------------------------------------------
// Dual-A fp32 WMMA GEMM: Out = act(Arel@Wrel + Aroot@Wroot + bias)
// A row-major [M,K], W row-major [K,Ncols], Out row-major [M,Ncols].
// One wave32 computes one 16x16 tile via V_WMMA_F32_16X16X4_F32, K/4 steps
// per operand pair, shared f32 accumulator. M%16==0, Ncols%16==0.
// ---------------------------------------------------------------------------
__global__ __launch_bounds__(256) void gemm_dual_kernel(
    const float* __restrict__ Arel, const float* __restrict__ Wrel,
    const float* __restrict__ Aroot, const float* __restrict__ Wroot,
    const float* __restrict__ bias, float* __restrict__ Out,
    int M, int K, int Ncols, int do_relu)
{
    int wave = (int)((blockIdx.x * blockDim.x + threadIdx.x) >> 5);
    int lane = threadIdx.x & 31;
    int ntiles = Ncols >> 4;
    int mtile = wave / ntiles;
    int ntile = wave - mtile * ntiles;
    int m0 = mtile << 4;
    int n0 = ntile << 4;
    if (m0 >= M) return;                         // wave-uniform: EXEC stays all-1s

    int half = lane >> 4;                        // 0: K even pair base, 1: K+2
    int l16  = lane & 15;

    v8f c = {};
    const float* arow = Arel + (size_t)(m0 + l16) * K + 2 * half;
    for (int k0 = 0; k0 < K; k0 += 4) {
        v2f a; a.x = arow[k0]; a.y = arow[k0 + 1];
        const float* bp = Wrel + (size_t)(k0 + 2 * half) * Ncols + n0 + l16;
        v2f b; b.x = bp[0]; b.y = bp[Ncols];
        c = __builtin_amdgcn_wmma_f32_16x16x4_f32(false, a, false, b,
                                                  (short)0, c, false, false);
    }
    arow = Aroot + (size_t)(m0 + l16) * K + 2 * half;
    for (int k0 = 0; k0 < K; k0 += 4) {
        v2f a; a.x = arow[k0]; a.y = arow[k0 + 1];
        const float* bp = Wroot + (size_t)(k0 + 2 * half) * Ncols + n0 + l16;
        v2f b; b.x = bp[0]; b.y = bp[Ncols];
        c = __builtin_amdgcn_wmma_f32_16x16x4_f32(false, a, false, b,
                                                  (short)0, c, false, false);
    }

    float bv = bias[n0 + l16];
    int rbase = m0 + 8 * half;                   // C/D layout: lanes16-31 hold M=8..15
#pragma unroll
    for (int r = 0; r < 8; ++r) {
        float v = c[r] + bv;
        if (do_relu) v = fmaxf(v, 0.0f);
        Out[(size_t)(rbase + r) * Ncols + n0 + l16] = v;
    }
}

// ---------------------------------------------------------------------------
// Final linear: out[N,51] = [x1|x2] @ lin_w + lin_b.  K=512 (256 from each),
// N padded 51->64; out-of-range B columns are zero-masked (branch-free so
// EXEC is all-1s at every WMMA), stores guarded per-lane after the loop.
// ---------------------------------------------------------------------------
__global__ __launch_bounds__(256) void final_linear_kernel(
    const float* __restrict__ x1, const float* __restrict__ x2,
    const float* __restrict__ lw, const float* __restrict__ lb,
    float* __restrict__ Out)
{
    const int M = NN, Kh = HID, NC = OUT_C;
    int wave = (int)((blockIdx.x * blockDim.x + threadIdx.x) >> 5);
    int lane = threadIdx.x & 31;
    const int ntiles = 4;                        // 64 padded cols
    int mtile = wave / ntiles;
    int ntile = wave - mtile * ntiles;
    int m0 = mtile << 4;
    int n0 = ntile << 4;
    if (m0 >= M) return;

    int half = lane >> 4;
    int l16  = lane & 15;
    int col  = n0 + l16;
    int colc = col < NC ? col : NC - 1;          // clamped address
    float cmask = col < NC ? 1.0f : 0.0f;        // branch-free zero pad

    v8f c = {};
    const float* arow = x1 + (size_t)(m0 + l16) * Kh + 2 * half;
    for (int k0 = 0; k0 < Kh; k0 += 4) {
        v2f a; a.x = arow[k0]; a.y = arow[k0 + 1];
        const float* bp = lw + (size_t)(k0 + 2 * half) * NC + colc;
        v2f b; b.x = bp[0] * cmask; b.y = bp[NC] * cmask;
        c = __builtin_amdgcn_wmma_f32_16x16x4_f32(false, a, false, b,
                                                  (short)0, c, false, false);
    }
    arow = x2 + (size_t)(m0 + l16) * Kh + 2 * half;
    for (int k0 = 0; k0 < Kh; k0 += 4) {
        v2f a; a.x = arow[k0]; a.y = arow[k0 + 1];
        const float* bp = lw + (size_t)(Kh + k0 + 2 * half) * NC + colc;
        v2f b; b.x = bp[0] * cmask; b.y = bp[NC] * cmask;
        c = __builtin_amdgcn_wmma_f32_16x16x4_f32(false, a, false, b,
                                                  (short)0, c, false, false);
    }

    float bv = lb[colc];
    int rbase = m0 + 8 * half;
    if (col < NC) {
#pragma unroll
        for (int r = 0; r < 8; ++r)
            Out[(size_t)(rbase + r) * NC + col] = c[r] + bv;
    }
}

// ---------------------------------------------------------------------------
extern "C" void kernel_launch(void* const* d_in, const int* in_sizes, int n_in,
                              void* d_out, int out_size, void* d_ws, size_t ws_size,
                              hipStream_t stream) {
    const float*      x0      = (const float*)d_in[0];
    const long long*  ei      = (const long long*)d_in[1];   // int64 [2,E]
    const float*      ew      = (const float*)d_in[2];
    const float*      w_rel1  = (const float*)d_in[3];
    const float*      b_rel1  = (const float*)d_in[4];
    const float*      w_root1 = (const float*)d_in[5];
    const float*      w_rel2  = (const float*)d_in[6];
    const float*      b_rel2  = (const float*)d_in[7];
    const float*      w_root2 = (const float*)d_in[8];
    const float*      lin_w   = (const float*)d_in[9];
    const float*      lin_b   = (const float*)d_in[10];
    float*            out     = (float*)d_out;

    // Workspace: x1 | x2 | agg (reused for both layers), 3 * N * HID floats.
    float* x1  = (float*)d_ws;
    float* x2  = x1 + (size_t)NN * HID;
    float* agg = x2 + (size_t)NN * HID;

    const int B = 256;
    auto cdiv = [](long long a, long long b) { return (int)((a + b - 1) / b); };

    // ---- Layer 1 ----
    {
        long long n = (long long)NN * IN_C;
        fill_zero_kernel<<<cdiv(n, B), B, 0, stream>>>(agg, n);
        long long total = (long long)EE * (IN_C / 4);
        scatter_add_kernel<<<cdiv(total, B), B, 0, stream>>>(x0, ei, ew, agg, EE, IN_C);
        long long waves = ((long long)NN / 16) * (HID / 16);   // 50000 waves
        gemm_dual_kernel<<<cdiv(waves * 32, B), B, 0, stream>>>(
            agg, w_rel1, x0, w_root1, b_rel1, x1, NN, IN_C, HID, 1);
    }
    // ---- Layer 2 ----
    {
        long long n = (long long)NN * HID;
        fill_zero_kernel<<<cdiv(n, B), B, 0, stream>>>(agg, n);
        long long total = (long long)EE * (HID / 4);
        scatter_add_kernel<<<cdiv(total, B), B, 0, stream>>>(x1, ei, ew, agg, EE, HID);
        long long waves = ((long long)NN / 16) * (HID / 16);
        gemm_dual_kernel<<<cdiv(waves * 32, B), B, 0, stream>>>(
            agg, w_rel2, x1, w_root2, b_rel2, x2, NN, HID, HID, 1);
    }
    // ---- Final linear ----
    {
        long long waves = ((long long)NN / 16) * 4;            // 12500 waves
        final_linear_kernel<<<cdiv(waves * 32, B), B, 0, stream>>>(
            x1, x2, lin_w, lin_b, out);
    }
}